// NCELoss_43387759624411
// MI455X (gfx1250) — compile-verified
//
#include <hip/hip_runtime.h>

#define NROWS 8192
#define DDIM 128
#define NUM_CLASSES 100
#define INV_TEMP 10.0f
#define NCHUNK 8
#define COLS_PER_CHUNK (NROWS / NCHUNK) // 1024

typedef float v2f __attribute__((ext_vector_type(2)));
typedef float v8f __attribute__((ext_vector_type(8)));

// One wave (32 lanes) L2-normalizes one row of 128 floats.
__global__ void nce_normalize_kernel(const float* __restrict__ projs,
                                     float* __restrict__ reps) {
    const int wave = threadIdx.x >> 5;
    const int lane = threadIdx.x & 31;
    const int row  = blockIdx.x * 8 + wave;

    float4 v = ((const float4*)(projs + (size_t)row * DDIM))[lane];
    float s = v.x * v.x + v.y * v.y + v.z * v.z + v.w * v.w;
#pragma unroll
    for (int m = 16; m >= 1; m >>= 1) s += __shfl_xor(s, m, 32);
    const float r = 1.0f / sqrtf(s);
    float4 o = make_float4(v.x * r, v.y * r, v.z * r, v.w * r);
    ((float4*)(reps + (size_t)row * DDIM))[lane] = o;
}

// Each wave owns a 16-row strip; iterates over one 1024-column chunk in
// 16-wide tiles computed via V_WMMA_F32_16X16X4_F32 (K=128 => 32 WMMAs/tile).
// A strip is register-resident; B tile is one batched 32x b64 burst; next
// tile is prefetched near-scope. __launch_bounds__(256,2) gives the ~190
// VGPRs this needs without spilling.
__global__ void __launch_bounds__(256, 2)
nce_simloss_kernel(const float* __restrict__ reps,
                   const int* __restrict__ labels,
                   float* __restrict__ nom_part,
                   float* __restrict__ den_part) {
    const int wave = threadIdx.x >> 5;
    const int lane = threadIdx.x & 31;
    const int half = lane >> 4;   // 0: lanes 0-15, 1: lanes 16-31
    const int l16  = lane & 15;
    const int i0   = (blockIdx.x * 8 + wave) * 16;   // strip base row
    const int c0   = blockIdx.y * COLS_PER_CHUNK;    // column chunk base

    // A fragments: 16x4 f32 layout -> lane holds row (i0+l16),
    // VGPR pair covers K = 4t + 2*half + {0,1}. 32 frags = all of K=128.
    v2f a[32];
    const float* aptr = reps + (size_t)(i0 + l16) * DDIM + 2 * half;
#pragma unroll
    for (int t = 0; t < 32; ++t) a[t] = *(const v2f*)(aptr + t * 4);

    // Labels of my strip rows for each C VGPR v: row = i0 + v + 8*half.
    int labi[8];
#pragma unroll
    for (int v = 0; v < 8; ++v) labi[v] = labels[i0 + v + 8 * half];

    float nomAcc[8], denAcc[8];
#pragma unroll
    for (int v = 0; v < 8; ++v) { nomAcc[v] = 0.0f; denAcc[v] = 0.0f; }

    // Prefetch the first B tile of this chunk (64 x 128B lines via 2/lane).
    {
        const char* p0 = (const char*)(reps + (size_t)(c0 + l16) * DDIM) + half * 256;
        __builtin_prefetch(p0, 0, 3);
        __builtin_prefetch(p0 + 128, 0, 3);
    }

    for (int j0 = c0; j0 < c0 + COLS_PER_CHUNK; j0 += 16) {
        // Batched load of the whole 8KB B tile (32 outstanding b64 loads).
        const float* bptr = reps + (size_t)(j0 + l16) * DDIM + 2 * half;
        v2f b[32];
#pragma unroll
        for (int t = 0; t < 32; ++t) b[t] = *(const v2f*)(bptr + t * 4);
        const int labj = labels[j0 + l16];

        // Prefetch next tile while this tile's WMMA + exp work runs.
        if (j0 + 16 < c0 + COLS_PER_CHUNK) {
            const char* pn =
                (const char*)(reps + (size_t)(j0 + 16 + l16) * DDIM) + half * 256;
            __builtin_prefetch(pn, 0, 3);
            __builtin_prefetch(pn + 128, 0, 3);
        }

        v8f c = {0.0f, 0.0f, 0.0f, 0.0f, 0.0f, 0.0f, 0.0f, 0.0f};
#pragma unroll
        for (int t = 0; t < 32; ++t) {
            c = __builtin_amdgcn_wmma_f32_16x16x4_f32(
                    false, a[t], false, b[t], (short)0, c, false, false);
        }

        const int j = j0 + l16;   // C/D layout: N = lane % 16
#pragma unroll
        for (int v = 0; v < 8; ++v) {
            const int i = i0 + v + 8 * half;   // M = v + 8*(lane/16)
            float e = __expf(c[v] * INV_TEMP);
            e = (i == j) ? 0.0f : e;           // mask diagonal
            denAcc[v] += e;
            nomAcc[v] += (labi[v] == labj) ? e : 0.0f;
        }
    }

    // Reduce across the 16 lanes that share each row (within each half).
#pragma unroll
    for (int v = 0; v < 8; ++v) {
        float dsum = denAcc[v];
        float nsum = nomAcc[v];
#pragma unroll
        for (int m = 8; m >= 1; m >>= 1) {
            dsum += __shfl_xor(dsum, m, 16);
            nsum += __shfl_xor(nsum, m, 16);
        }
        if (l16 == 0) {
            const int row = i0 + v + 8 * half;
            den_part[blockIdx.y * NROWS + row] = dsum;
            nom_part[blockIdx.y * NROWS + row] = nsum;
        }
    }
}

// Single block: label histogram (deterministic int LDS atomics), combine chunk
// partials, per-row loss, fixed-order tree reduction -> mean.
__global__ void nce_finalize_kernel(const float* __restrict__ nom_part,
                                    const float* __restrict__ den_part,
                                    const int* __restrict__ labels,
                                    float* __restrict__ out) {
    __shared__ int   cnt[NUM_CLASSES];
    __shared__ float red[1024];
    const int tid = threadIdx.x;

    for (int c = tid; c < NUM_CLASSES; c += 1024) cnt[c] = 0;
    __syncthreads();
    for (int r = tid; r < NROWS; r += 1024) atomicAdd(&cnt[labels[r]], 1);
    __syncthreads();

    float acc = 0.0f;
    for (int r = tid; r < NROWS; r += 1024) {
        float nom = 0.0f, den = 0.0f;
#pragma unroll
        for (int ch = 0; ch < NCHUNK; ++ch) {
            nom += nom_part[ch * NROWS + r];
            den += den_part[ch * NROWS + r];
        }
        const int c = cnt[labels[r]] - 1;   // positives excluding self
        float loss = 0.0f;                  // rows w/o positives: log(1) = 0
        if (c > 0) {
            nom /= (float)c;
            loss = -logf(nom / den);
        }
        acc += loss;
    }
    red[tid] = acc;
    __syncthreads();
#pragma unroll
    for (int s = 512; s > 0; s >>= 1) {
        if (tid < s) red[tid] += red[tid + s];
        __syncthreads();
    }
    if (tid == 0) out[0] = red[0] / (float)NROWS;
}

extern "C" void kernel_launch(void* const* d_in, const int* in_sizes, int n_in,
                              void* d_out, int out_size, void* d_ws, size_t ws_size,
                              hipStream_t stream) {
    (void)in_sizes; (void)n_in; (void)out_size; (void)ws_size;
    const float* projs  = (const float*)d_in[0];
    const int*   labels = (const int*)d_in[1];
    float*       out    = (float*)d_out;

    // Workspace layout: reps[8192*128] | nom_part[8*8192] | den_part[8*8192]
    float* reps     = (float*)d_ws;
    float* nom_part = reps + (size_t)NROWS * DDIM;
    float* den_part = nom_part + (size_t)NCHUNK * NROWS;

    nce_normalize_kernel<<<NROWS / 8, 256, 0, stream>>>(projs, reps);
    nce_simloss_kernel<<<dim3(NROWS / 128, NCHUNK), 256, 0, stream>>>(
        reps, labels, nom_part, den_part);
    nce_finalize_kernel<<<1, 1024, 0, stream>>>(nom_part, den_part, labels, out);
}